// SelectGroupFClayer_38757784879389
// MI455X (gfx1250) — compile-verified
//
#include <hip/hip_runtime.h>
#include <math.h>

// ---------------------------------------------------------------------------
// SelectGroupFClayer for MI455X (gfx1250, wave32).
//
// Roofline: matmul work is ~0.5 GFLOP (negligible); HBM traffic dominates:
// 256MB read (reduce) + 512MB read/write (final scale) => ~33us @ 23.3 TB/s.
// Middle stage uses V_WMMA_F32_16X16X4_F32 (full f32 precision) on 16x16
// output tiles; each 16-wide channel tile sits inside one group for all
// g in {2,4,8,16}, so the block-diagonal GEMM maps exactly onto WMMA tiles.
//
//   K1 reduce : y[b,i] = mean_j(x[b, i*512+j]) + max_j(...)
//   K2 selcoef: coef1[b,0..3] via softmax-select over y@Wsel1^T
//   K3 branch : temp = sum_m coef1[.,m] * relu(gconv_m(y)+b_m)     (WMMA)
//   K4 selcoef: coef2 over temp@Wsel2^T
//   K5 branch : att  = sum_m coef2[.,m] * sigmoid(gconv_m(temp)+b_m)
//   K6 scale  : out[b,c,hw] = x[b,c,hw] * att[b,hw]
// ---------------------------------------------------------------------------

typedef __attribute__((ext_vector_type(2))) float v2f;
typedef __attribute__((ext_vector_type(8))) float v8f;

#define BATCH 128
#define CDIM  1024   // channel dim of y/temp/att
#define ROWLEN 512   // contiguous reduction length

// ---- K1: per-row (512 contiguous floats) mean+max, one wave per row -------
__global__ __launch_bounds__(256) void k_reduce(const float* __restrict__ x,
                                                float* __restrict__ y) {
  const int row  = blockIdx.x * 8 + (threadIdx.x >> 5);   // 131072 rows
  const int lane = threadIdx.x & 31;
  const float4* xp = (const float4*)(x + (size_t)row * ROWLEN);
  float s = 0.f, m = -__builtin_inff();
#pragma unroll
  for (int it = 0; it < 4; ++it) {                        // 16 floats per lane
    float4 v = xp[lane + it * 32];                        // coalesced B128
    s += v.x + v.y + v.z + v.w;
    m = fmaxf(m, fmaxf(fmaxf(v.x, v.y), fmaxf(v.z, v.w)));
  }
#pragma unroll
  for (int off = 16; off > 0; off >>= 1) {
    s += __shfl_xor(s, off, 32);
    m = fmaxf(m, __shfl_xor(m, off, 32));
  }
  if (lane == 0) y[row] = s * (1.0f / 512.0f) + m;
}

// ---- K2/K4: selection coefficients, one wave per batch ---------------------
__global__ __launch_bounds__(32) void k_selcoef(const float* __restrict__ vin,
                                                const float* __restrict__ Wsel,
                                                float* __restrict__ coef) {
  const int b = blockIdx.x;
  const int lane = threadIdx.x;
  float a0 = 0.f, a1 = 0.f, a2 = 0.f, a3 = 0.f;
  for (int c = lane; c < CDIM; c += 32) {
    float yv = vin[b * CDIM + c];
    a0 += yv * Wsel[c];
    a1 += yv * Wsel[CDIM + c];
    a2 += yv * Wsel[2 * CDIM + c];
    a3 += yv * Wsel[3 * CDIM + c];
  }
#pragma unroll
  for (int off = 16; off > 0; off >>= 1) {
    a0 += __shfl_xor(a0, off, 32);
    a1 += __shfl_xor(a1, off, 32);
    a2 += __shfl_xor(a2, off, 32);
    a3 += __shfl_xor(a3, off, 32);
  }
  if (lane == 0) {
    float o[4] = {a0, a1, a2, a3};
    float mn = fminf(fminf(o[0], o[1]), fminf(o[2], o[3]));
#pragma unroll
    for (int n = 0; n < 4; ++n) o[n] = o[n] - mn + 1e-15f;
    float mx = fmaxf(fmaxf(o[0], o[1]), fmaxf(o[2], o[3]));
    float lo[4], hi[4], wl[4], wh[4], s = 0.f;
#pragma unroll
    for (int n = 0; n < 4; ++n) {
      o[n] = o[n] / mx * 3.0f;
      lo[n] = floorf(o[n]);
      hi[n] = ceilf(o[n]);
      wl[n] = expf(o[n] - lo[n]);   // joint softmax over 8 entries
      wh[n] = expf(hi[n] - o[n]);
      s += wl[n] + wh[n];
    }
    float c4[4] = {0.f, 0.f, 0.f, 0.f};
#pragma unroll
    for (int n = 0; n < 4; ++n) {
      int il = min(3, max(0, (int)lo[n]));
      int ih = min(3, max(0, (int)hi[n]));
      c4[il] += wl[n] / s;
      c4[ih] += wh[n] / s;
    }
#pragma unroll
    for (int m = 0; m < 4; ++m) coef[b * 4 + m] = c4[m];
  }
}

// ---- per-branch WMMA accumulation, CG = group width (compile-time) ---------
// Direct __restrict__ global pointers (no pointer array) so the compiler
// keeps loads in the global address space (global_load_b64, not flat_load).
template <int CG>
__device__ __forceinline__ void branch_accum(
    const float* __restrict__ vin, const float* __restrict__ W,
    const float* __restrict__ bv, const float* __restrict__ coef,
    int mt, int colBase, int lm, int half, int nb, int use_sigmoid,
    float out8[8]) {
  const int go = colBase / CG;               // compile-time strength-reduced
  // A rows: vin[batch = mt*16+lm, group-input channels]; K = k0 + v + 2*half
  const float* arow = vin + (size_t)(mt * 16 + lm) * CDIM + go * CG;
  // B rows: W[go, o_local, k] with o_local = colBase - go*CG + lm
  const float* brow = W + (size_t)go * CG * CG
                    + (size_t)(colBase - go * CG + lm) * CG;
  const int koff = 2 * half;
  v8f acc = {0.f, 0.f, 0.f, 0.f, 0.f, 0.f, 0.f, 0.f};
  for (int k0 = 0; k0 < CG; k0 += 4) {
    v2f a, bf;
    a.x = arow[k0 + koff + 0];
    a.y = arow[k0 + koff + 1];
    bf.x = brow[k0 + koff + 0];
    bf.y = brow[k0 + koff + 1];
    acc = __builtin_amdgcn_wmma_f32_16x16x4_f32(
        false, a, false, bf, (short)0, acc, false, false);
  }
  const float bias = bv[colBase + lm];       // per-column bias
#pragma unroll
  for (int v = 0; v < 8; ++v) {              // C/D: M = v + 8*half, N = lm
    const int batch = mt * 16 + v + 8 * half;
    float val = acc[v] + bias;
    val = use_sigmoid ? (1.0f / (1.0f + expf(-val))) : fmaxf(val, 0.f);
    out8[v] += coef[batch * 4 + nb] * val;
  }
}

// ---- K3/K5: fused 4x grouped-GEMM + activation + coef combine --------------
// One wave per 16x16 output tile (8 batch-tiles x 64 channel-tiles).
__global__ __launch_bounds__(32) void k_branch(
    const float* __restrict__ vin,                    // (128,1024)
    const float* __restrict__ Wg0, const float* __restrict__ Wg1,
    const float* __restrict__ Wg2, const float* __restrict__ Wg3,
    const float* __restrict__ bv0, const float* __restrict__ bv1,
    const float* __restrict__ bv2, const float* __restrict__ bv3,
    const float* __restrict__ coef,                   // (128,4)
    float* __restrict__ vout,                         // (128,1024)
    int use_sigmoid) {
  const int tile = blockIdx.x;        // 512 tiles
  const int mt = tile >> 6;           // batch tile 0..7
  const int nt = tile & 63;           // channel tile 0..63
  const int lane = threadIdx.x;
  const int half = lane >> 4;         // 0/1
  const int lm = lane & 15;
  const int colBase = nt * 16;

  float out8[8];
#pragma unroll
  for (int v = 0; v < 8; ++v) out8[v] = 0.f;

  branch_accum<512>(vin, Wg0, bv0, coef, mt, colBase, lm, half, 0, use_sigmoid, out8);
  branch_accum<256>(vin, Wg1, bv1, coef, mt, colBase, lm, half, 1, use_sigmoid, out8);
  branch_accum<128>(vin, Wg2, bv2, coef, mt, colBase, lm, half, 2, use_sigmoid, out8);
  branch_accum< 64>(vin, Wg3, bv3, coef, mt, colBase, lm, half, 3, use_sigmoid, out8);

#pragma unroll
  for (int v = 0; v < 8; ++v) {
    const int batch = mt * 16 + v + 8 * half;
    vout[(size_t)batch * CDIM + colBase + lm] = out8[v];
  }
}

// ---- K6: out = x * att broadcast over 512 channels, float4 streams ---------
__global__ __launch_bounds__(256) void k_scale(const float* __restrict__ x,
                                               const float* __restrict__ att,
                                               float* __restrict__ out) {
  const unsigned i4 = blockIdx.x * 256u + threadIdx.x;  // 16,777,216 float4s
  const float4 xv = ((const float4*)x)[i4];
  const unsigned b = i4 >> 17;                          // / (512*1024/4)
  const unsigned hw4 = i4 & 255u;                       // 1024/4 per row
  const float4 a = ((const float4*)att)[b * 256u + hw4];
  float4 o;
  o.x = xv.x * a.x; o.y = xv.y * a.y; o.z = xv.z * a.z; o.w = xv.w * a.w;
  ((float4*)out)[i4] = o;
}

extern "C" void kernel_launch(void* const* d_in, const int* in_sizes, int n_in,
                              void* d_out, int out_size, void* d_ws, size_t ws_size,
                              hipStream_t stream) {
  (void)in_sizes; (void)n_in; (void)out_size; (void)ws_size;
  const float* x  = (const float*)d_in[0];
  const float* W1 = (const float*)d_in[1];  const float* b1 = (const float*)d_in[2];
  const float* W2 = (const float*)d_in[3];  const float* b2 = (const float*)d_in[4];
  const float* W3 = (const float*)d_in[5];  const float* b3 = (const float*)d_in[6];
  const float* W4 = (const float*)d_in[7];  const float* b4 = (const float*)d_in[8];
  const float* W5 = (const float*)d_in[9];  const float* b5 = (const float*)d_in[10];
  const float* W6 = (const float*)d_in[11]; const float* b6 = (const float*)d_in[12];
  const float* W7 = (const float*)d_in[13]; const float* b7 = (const float*)d_in[14];
  const float* W8 = (const float*)d_in[15]; const float* b8 = (const float*)d_in[16];
  const float* Wsel1 = (const float*)d_in[17];
  const float* Wsel2 = (const float*)d_in[18];
  float* out = (float*)d_out;

  char* ws = (char*)d_ws;
  float* y     = (float*)(ws);                     // 128*1024 f32 = 512KB
  float* temp  = (float*)(ws + (1u << 19));        // 512KB
  float* att   = (float*)(ws + 2u * (1u << 19));   // 512KB
  float* coef1 = (float*)(ws + 3u * (1u << 19));   // 2KB
  float* coef2 = (float*)(ws + 3u * (1u << 19) + 4096);

  // K1: 131072 rows, 8 rows (waves) per 256-thread block
  k_reduce<<<16384, 256, 0, stream>>>(x, y);
  // K2: one wave per batch
  k_selcoef<<<BATCH, 32, 0, stream>>>(y, Wsel1, coef1);
  // K3: relu stage
  k_branch<<<512, 32, 0, stream>>>(y, W1, W2, W3, W4, b1, b2, b3, b4,
                                   coef1, temp, 0);
  // K4
  k_selcoef<<<BATCH, 32, 0, stream>>>(temp, Wsel2, coef2);
  // K5: sigmoid stage
  k_branch<<<512, 32, 0, stream>>>(temp, W5, W6, W7, W8, b5, b6, b7, b8,
                                   coef2, att, 1);
  // K6: 67,108,864 floats as 16,777,216 float4s
  k_scale<<<65536, 256, 0, stream>>>(x, att, out);
}